// StaticGCN_46445776339763
// MI455X (gfx1250) — compile-verified
//
#include <hip/hip_runtime.h>
#include <hip/hip_bf16.h>

#define N_NODES 131072
#define N_EDGES 1048576
#define D_IN 64
#define D_HID 64
#define D_OUT 3
#define NUM_GRAPHS 32
#define NODES_PER_GRAPH (N_NODES / NUM_GRAPHS)   // 4096

typedef __attribute__((ext_vector_type(2))) float v2f;
typedef __attribute__((ext_vector_type(8))) float v8f;

// ---------------------------------------------------------------------------
// Kernel 0: initialize workspace (agg1 = 0, agg2 = 0, deg = 1.0 (self loop),
// out = 0). Grid covers N_NODES*64 elements.
// ---------------------------------------------------------------------------
__global__ __launch_bounds__(256) void k_init(float* __restrict__ deg,
                                              float* __restrict__ agg1,
                                              float* __restrict__ agg2,
                                              float* __restrict__ out) {
  const int i = blockIdx.x * 256 + threadIdx.x;
  agg1[i] = 0.0f;
  if (i < N_NODES) deg[i] = 1.0f;           // +1 for self-loop
  if (i < N_NODES * D_OUT) agg2[i] = 0.0f;
  if (i < NUM_GRAPHS * D_OUT) out[i] = 0.0f;
}

// ---------------------------------------------------------------------------
// Kernel 1: degree at target nodes (col), hardware fp32 atomic add.
// ---------------------------------------------------------------------------
__global__ __launch_bounds__(256) void k_degree(const int* __restrict__ ei,
                                                float* __restrict__ deg) {
  const int e = blockIdx.x * 256 + threadIdx.x;
  const int c = ei[N_EDGES + e];
  unsafeAtomicAdd(&deg[c], 1.0f);
}

// ---------------------------------------------------------------------------
// Kernel 2: deg -> deg^{-1/2} in place.
// ---------------------------------------------------------------------------
__global__ __launch_bounds__(256) void k_rsqrt(float* __restrict__ deg) {
  const int i = blockIdx.x * 256 + threadIdx.x;
  deg[i] = rsqrtf(deg[i]);
}

// ---------------------------------------------------------------------------
// Kernel 3: h1 = x @ W1 using native fp32 WMMA (V_WMMA_F32_16X16X4_F32).
// One wave per 16x16 output tile; K=64 handled as 16 chained K=4 WMMAs.
// VGPR layouts per CDNA5 ISA 7.12.2:
//   A 16x4 : lanes 0-15 hold K=0,1 in v0,v1 ; lanes 16-31 hold K=2,3
//   B 4x16 : same K striping, N = lane&15 within a VGPR
//   C 16x16: VGPR r -> row (r or 8+r), col = lane&15
// Branches are wave-uniform so EXEC is all-1s at every WMMA (ISA requirement).
// ---------------------------------------------------------------------------
__global__ __launch_bounds__(128) void k_gemm1_wmma(const float* __restrict__ X,
                                                    const float* __restrict__ W,
                                                    float* __restrict__ H) {
  const int wave = blockIdx.x * (128 >> 5) + (threadIdx.x >> 5);
  const int lane = threadIdx.x & 31;
  const int rowTile = wave >> 2;          // N_NODES/16 row tiles
  const int colTile = wave & 3;           // 64/16 col tiles
  const int rowBase = rowTile << 4;
  const int colBase = colTile << 4;
  if (rowBase >= N_NODES) return;         // wave-uniform

  const int m    = lane & 15;
  const int half = lane >> 4;             // 0: K pair {0,1}, 1: K pair {2,3}
  const int koff = half << 1;

  const float* xrow = X + (size_t)(rowBase + m) * D_IN;        // A: row = m
  const float* wcol = W + colBase + m;                         // B: col = m

  v8f c = {0.f, 0.f, 0.f, 0.f, 0.f, 0.f, 0.f, 0.f};
#pragma unroll
  for (int k = 0; k < D_IN; k += 4) {
    v2f a, b;
    a.x = xrow[k + koff];
    a.y = xrow[k + koff + 1];
    b.x = wcol[(size_t)(k + koff) * D_HID];
    b.y = wcol[(size_t)(k + koff + 1) * D_HID];
    // 8 args: (neg_a, A, neg_b, B, c_mod, C, reuse_a, reuse_b)
    c = __builtin_amdgcn_wmma_f32_16x16x4_f32(false, a, false, b,
                                              (short)0, c, false, false);
  }

  // C/D layout: lanes 0-15 rows 0..7, lanes 16-31 rows 8..15
  float* hout = H + (size_t)(rowBase + (half << 3)) * D_HID + colBase + m;
#pragma unroll
  for (int r = 0; r < 8; ++r) hout[(size_t)r * D_HID] = c[r];
}

// ---------------------------------------------------------------------------
// Kernel 4: layer-1 message scatter. 32 lanes per edge, 2 dims per lane.
// h1 (32MB) and agg1 (32MB) both fit in the 192MB L2 -> L2-resident atomics.
// ---------------------------------------------------------------------------
__global__ __launch_bounds__(256) void k_scatter1(const int* __restrict__ ei,
                                                  const float* __restrict__ dis,
                                                  const float* __restrict__ h1,
                                                  float* __restrict__ agg1) {
  const unsigned t = blockIdx.x * 256 + threadIdx.x;
  const int e = (int)(t >> 5);
  const int lane = (int)(t & 31);
  const int r = ei[e];
  const int c = ei[N_EDGES + e];
  const float norm = dis[r] * dis[c];
  const float2 hv = *(const float2*)(h1 + (size_t)r * D_HID + lane * 2);
  float* dst = agg1 + (size_t)c * D_HID + lane * 2;
  unsafeAtomicAdd(dst + 0, hv.x * norm);
  unsafeAtomicAdd(dst + 1, hv.y * norm);
}

// ---------------------------------------------------------------------------
// Kernel 5: fused self-loop + bias + ReLU, in place on agg1 -> hrelu.
// ---------------------------------------------------------------------------
__global__ __launch_bounds__(256) void k_finalize1(const float* __restrict__ h1,
                                                   const float* __restrict__ dis,
                                                   const float* __restrict__ b1,
                                                   float* __restrict__ agg1) {
  const int i = blockIdx.x * 256 + threadIdx.x;
  const int node = i >> 6;
  const int d = i & 63;
  const float ds = dis[node];
  const float v = agg1[i] + h1[i] * (ds * ds) + b1[d];
  agg1[i] = fmaxf(v, 0.0f);
}

// ---------------------------------------------------------------------------
// Kernel 6: z = hrelu @ W2  (64 -> 3). One node per thread; W2 (768B) lives
// in scalar/vector cache. WMMA would waste 13/16 columns here.
// ---------------------------------------------------------------------------
__global__ __launch_bounds__(256) void k_gemm2(const float* __restrict__ hrelu,
                                               const float* __restrict__ W2,
                                               float* __restrict__ z) {
  const int n = blockIdx.x * 256 + threadIdx.x;
  const float* hr = hrelu + (size_t)n * D_HID;
  float a0 = 0.f, a1 = 0.f, a2 = 0.f;
#pragma unroll
  for (int k = 0; k < D_HID; ++k) {
    const float hv = hr[k];
    a0 = fmaf(hv, W2[k * 3 + 0], a0);
    a1 = fmaf(hv, W2[k * 3 + 1], a1);
    a2 = fmaf(hv, W2[k * 3 + 2], a2);
  }
  z[n * 3 + 0] = a0;
  z[n * 3 + 1] = a1;
  z[n * 3 + 2] = a2;
}

// ---------------------------------------------------------------------------
// Kernel 7: layer-2 message scatter (3 floats/edge). One edge per thread.
// ---------------------------------------------------------------------------
__global__ __launch_bounds__(256) void k_scatter2(const int* __restrict__ ei,
                                                  const float* __restrict__ dis,
                                                  const float* __restrict__ z,
                                                  float* __restrict__ agg2) {
  const int e = blockIdx.x * 256 + threadIdx.x;
  const int r = ei[e];
  const int c = ei[N_EDGES + e];
  const float norm = dis[r] * dis[c];
  const float* zs = z + (size_t)r * 3;
  float* dst = agg2 + (size_t)c * 3;
  unsafeAtomicAdd(dst + 0, zs[0] * norm);
  unsafeAtomicAdd(dst + 1, zs[1] * norm);
  unsafeAtomicAdd(dst + 2, zs[2] * norm);
}

// ---------------------------------------------------------------------------
// Kernel 8: fused layer-2 finalize + global mean pool. Each 256-thread block
// covers 256 contiguous nodes of one graph (4096 per graph); LDS tree-reduce
// then 3 atomics per block (1536 total instead of ~400K).
// ---------------------------------------------------------------------------
__global__ __launch_bounds__(256) void k_finalize_pool(const float* __restrict__ agg2,
                                                       const float* __restrict__ z,
                                                       const float* __restrict__ dis,
                                                       const float* __restrict__ b2,
                                                       const int* __restrict__ batch,
                                                       float* __restrict__ out) {
  __shared__ float s0[256], s1[256], s2[256];
  const int tid = threadIdx.x;
  const int n = blockIdx.x * 256 + tid;
  const float ds = dis[n];
  const float d2 = ds * ds;
  s0[tid] = agg2[n * 3 + 0] + z[n * 3 + 0] * d2 + b2[0];
  s1[tid] = agg2[n * 3 + 1] + z[n * 3 + 1] * d2 + b2[1];
  s2[tid] = agg2[n * 3 + 2] + z[n * 3 + 2] * d2 + b2[2];
  __syncthreads();
#pragma unroll
  for (int off = 128; off > 0; off >>= 1) {
    if (tid < off) {
      s0[tid] += s0[tid + off];
      s1[tid] += s1[tid + off];
      s2[tid] += s2[tid + off];
    }
    __syncthreads();
  }
  if (tid == 0) {
    const int g = batch[blockIdx.x * 256];  // block-uniform (contiguous batch)
    const float scale = 1.0f / (float)NODES_PER_GRAPH;
    unsafeAtomicAdd(out + g * 3 + 0, s0[0] * scale);
    unsafeAtomicAdd(out + g * 3 + 1, s1[0] * scale);
    unsafeAtomicAdd(out + g * 3 + 2, s2[0] * scale);
  }
}

// ---------------------------------------------------------------------------
// Host-side launcher. Inputs: x, edge_index, batch, W1, b1, W2, b2.
// Workspace layout (floats): [deg/dis N][h1 N*64][agg1 N*64][z N*3][agg2 N*3]
//   = N*135 floats ~= 68 MB.
// ---------------------------------------------------------------------------
extern "C" void kernel_launch(void* const* d_in, const int* in_sizes, int n_in,
                              void* d_out, int out_size, void* d_ws, size_t ws_size,
                              hipStream_t stream) {
  const float* x  = (const float*)d_in[0];
  const int* ei   = (const int*)d_in[1];
  const int* batch = (const int*)d_in[2];
  const float* W1 = (const float*)d_in[3];
  const float* b1 = (const float*)d_in[4];
  const float* W2 = (const float*)d_in[5];
  const float* b2 = (const float*)d_in[6];
  float* out = (float*)d_out;

  float* ws   = (float*)d_ws;
  float* deg  = ws;                              // N  (becomes deg^{-1/2})
  float* h1   = deg + N_NODES;                   // N*64
  float* agg1 = h1 + (size_t)N_NODES * D_HID;    // N*64 (becomes hrelu)
  float* z    = agg1 + (size_t)N_NODES * D_HID;  // N*3
  float* agg2 = z + (size_t)N_NODES * D_OUT;     // N*3

  // 0) init
  k_init<<<(N_NODES * D_HID) / 256, 256, 0, stream>>>(deg, agg1, agg2, out);
  // 1) degree (col targets), then 2) rsqrt in place
  k_degree<<<N_EDGES / 256, 256, 0, stream>>>(ei, deg);
  k_rsqrt<<<N_NODES / 256, 256, 0, stream>>>(deg);
  // 3) h1 = x @ W1   (WMMA fp32, one wave per 16x16 tile, 4 waves/block)
  k_gemm1_wmma<<<(N_NODES / 16) * (D_HID / 16) / 4, 128, 0, stream>>>(x, W1, h1);
  // 4) edge scatter layer 1 (32 lanes/edge)
  k_scatter1<<<(N_EDGES * 32) / 256, 256, 0, stream>>>(ei, deg, h1, agg1);
  // 5) self-loop + bias + relu (in place -> hrelu)
  k_finalize1<<<(N_NODES * D_HID) / 256, 256, 0, stream>>>(h1, deg, b1, agg1);
  // 6) z = hrelu @ W2
  k_gemm2<<<N_NODES / 256, 256, 0, stream>>>(agg1, W2, z);
  // 7) edge scatter layer 2 (1 thread/edge, 3 dims)
  k_scatter2<<<N_EDGES / 256, 256, 0, stream>>>(ei, deg, z, agg2);
  // 8) finalize + mean pool
  k_finalize_pool<<<N_NODES / 256, 256, 0, stream>>>(agg2, z, deg, b2, batch, out);
}